// PropagateUnit_39067022524699
// MI455X (gfx1250) — compile-verified
//
#include <hip/hip_runtime.h>
#include <hip/hip_bf16.h>

#define N_USERS 100000
#define N_ITEMS 50000
#define N_NODES 150000
#define DIM 32
#define N_EDGES 2400000
#define K_STEPS 10

// ---------------------------------------------------------------------------
// CDNA5 async LDS copy helpers (gfx1250: GLOBAL_LOAD_ASYNC_TO_LDS_B64, ASYNCcnt)
// ---------------------------------------------------------------------------
#if defined(__has_builtin)
#if __has_builtin(__builtin_amdgcn_global_load_async_to_lds_b64)
#define HAVE_ASYNC_LDS 1
#endif
#endif
#ifndef HAVE_ASYNC_LDS
#define HAVE_ASYNC_LDS 0
#endif

typedef int v2i __attribute__((ext_vector_type(2)));

__device__ __forceinline__ void async_copy_edge_b64(const void* g, void* l) {
#if HAVE_ASYNC_LDS
    __builtin_amdgcn_global_load_async_to_lds_b64(
        (__attribute__((address_space(1))) v2i*)(g),
        (__attribute__((address_space(3))) v2i*)(l),
        0, 0);
#else
    *(int2*)l = *(const int2*)g;   // synchronous fallback
#endif
}

__device__ __forceinline__ void wait_async_lds() {
#if HAVE_ASYNC_LDS
#if __has_builtin(__builtin_amdgcn_s_wait_asynccnt)
    __builtin_amdgcn_s_wait_asynccnt(0);
#else
    asm volatile("s_wait_asynccnt 0" ::: "memory");
#endif
#endif
}

// ---------------------------------------------------------------------------
// 1) max row-norm^2 of concat(xu, xi): wave32 shuffle reduce + bitwise atomicMax
// ---------------------------------------------------------------------------
__global__ __launch_bounds__(256) void norm_kernel(const float* __restrict__ xu,
                                                   const float* __restrict__ xi,
                                                   unsigned* __restrict__ maxsq_bits) {
    const int lane = threadIdx.x & 31;
    const int wid  = threadIdx.x >> 5;
    const int row  = blockIdx.x * 8 + wid;
    float v = 0.0f;
    if (row < N_NODES) {
        v = (row < N_USERS) ? xu[row * DIM + lane]
                            : xi[(row - N_USERS) * DIM + lane];
    }
    float s = v * v;
#pragma unroll
    for (int off = 16; off > 0; off >>= 1) s += __shfl_xor(s, off, 32);
    __shared__ float smax[8];
    if (lane == 0) smax[wid] = s;
    __syncthreads();
    if (threadIdx.x == 0) {
        float m = smax[0];
#pragma unroll
        for (int i = 1; i < 8; ++i) m = fmaxf(m, smax[i]);
        atomicMax(maxsq_bits, __float_as_uint(m));   // valid: all values >= 0
    }
}

// ---------------------------------------------------------------------------
// 2) CSR build: histogram -> scan -> scatter fill of packed (src, w) records
// ---------------------------------------------------------------------------
__global__ __launch_bounds__(256) void count_kernel(const int* __restrict__ dst,
                                                    int* __restrict__ counts) {
    int e = blockIdx.x * 256 + threadIdx.x;
    if (e < N_EDGES) atomicAdd(&counts[dst[e]], 1);
}

__global__ __launch_bounds__(256) void scan1_kernel(const int* __restrict__ counts,
                                                    int* __restrict__ row_ptr,
                                                    int* __restrict__ partial, int n) {
    __shared__ int sh[256];
    const int tid = threadIdx.x;
    const int i = blockIdx.x * 256 + tid;
    int v = (i < n) ? counts[i] : 0;
    sh[tid] = v;
    __syncthreads();
    for (int off = 1; off < 256; off <<= 1) {
        int t = (tid >= off) ? sh[tid - off] : 0;
        __syncthreads();
        sh[tid] += t;
        __syncthreads();
    }
    if (i < n) row_ptr[i] = sh[tid] - v;          // block-local exclusive
    if (tid == 255) partial[blockIdx.x] = sh[255]; // block total
}

__global__ __launch_bounds__(1024) void scan2_kernel(int* __restrict__ partial, int np) {
    __shared__ int sh[1024];
    const int tid = threadIdx.x;
    int v = (tid < np) ? partial[tid] : 0;
    sh[tid] = v;
    __syncthreads();
    for (int off = 1; off < 1024; off <<= 1) {
        int t = (tid >= off) ? sh[tid - off] : 0;
        __syncthreads();
        sh[tid] += t;
        __syncthreads();
    }
    if (tid < np) partial[tid] = sh[tid] - v;     // exclusive block offsets
}

__global__ __launch_bounds__(256) void scan3_kernel(int* __restrict__ row_ptr,
                                                    int* __restrict__ cursor,
                                                    const int* __restrict__ partial, int n) {
    int i = blockIdx.x * 256 + threadIdx.x;
    if (i < n) {
        int v = row_ptr[i] + partial[blockIdx.x];
        row_ptr[i] = v;
        cursor[i]  = v;
    }
    if (i == 0) row_ptr[n] = N_EDGES;
}

__global__ __launch_bounds__(256) void fill_kernel(const int* __restrict__ src,
                                                   const int* __restrict__ dst,
                                                   const float* __restrict__ w,
                                                   int* __restrict__ cursor,
                                                   int2* __restrict__ ep) {
    int e = blockIdx.x * 256 + threadIdx.x;
    if (e >= N_EDGES) return;
    int d = dst[e];
    int pos = atomicAdd(&cursor[d], 1);
    ep[pos] = make_int2(src[e], __float_as_int(w[e]));
}

// ---------------------------------------------------------------------------
// 3) h = concat(xu, xi) / norm   (float4 vectorized)
// ---------------------------------------------------------------------------
__global__ __launch_bounds__(256) void init_h_kernel(const float4* __restrict__ xu4,
                                                     const float4* __restrict__ xi4,
                                                     float4* __restrict__ h4,
                                                     const unsigned* __restrict__ maxsq_bits) {
    int i = blockIdx.x * 256 + threadIdx.x;          // over N_NODES * (DIM/4)
    if (i >= N_NODES * (DIM / 4)) return;
    const float inv = 1.0f / sqrtf(__uint_as_float(*maxsq_bits));
    float4 v = (i < N_USERS * (DIM / 4)) ? xu4[i] : xi4[i - N_USERS * (DIM / 4)];
    float4 r;
    r.x = v.x * inv; r.y = v.y * inv; r.z = v.z * inv; r.w = v.w * inv;
    h4[i] = r;
}

// ---------------------------------------------------------------------------
// 4) CSR propagation: one wave per destination row, lane = dim.
//    Edge records streamed into LDS with gfx1250 async-to-LDS + s_wait_asynccnt.
// ---------------------------------------------------------------------------
__global__ __launch_bounds__(256) void prop_kernel(const float* __restrict__ in,
                                                   float* __restrict__ out,
                                                   const int2* __restrict__ ep,
                                                   const int* __restrict__ rp) {
    __shared__ int2 sh_e[8][32];                     // per-wave staging, 2 KB
    const int lane = threadIdx.x & 31;
    const int wid  = threadIdx.x >> 5;
    const int row  = blockIdx.x * 8 + wid;
    if (row >= N_NODES) return;
    const int beg = rp[row];
    const int end = rp[row + 1];
    float acc = 0.0f;
    for (int base = beg; base < end; base += 32) {
        const int cnt = min(32, end - base);
        if (lane < cnt) {
            async_copy_edge_b64(&ep[base + lane], &sh_e[wid][lane]);
        }
        wait_async_lds();                            // ASYNCcnt == 0 before LDS reads
        int j = 0;
        for (; j + 4 <= cnt; j += 4) {               // 4 gathers in flight
            int2 e0 = sh_e[wid][j + 0];
            int2 e1 = sh_e[wid][j + 1];
            int2 e2 = sh_e[wid][j + 2];
            int2 e3 = sh_e[wid][j + 3];
            float v0 = in[e0.x * DIM + lane];
            float v1 = in[e1.x * DIM + lane];
            float v2 = in[e2.x * DIM + lane];
            float v3 = in[e3.x * DIM + lane];
            acc = fmaf(__int_as_float(e0.y), v0, acc);
            acc = fmaf(__int_as_float(e1.y), v1, acc);
            acc = fmaf(__int_as_float(e2.y), v2, acc);
            acc = fmaf(__int_as_float(e3.y), v3, acc);
        }
        for (; j < cnt; ++j) {
            int2 e = sh_e[wid][j];
            acc = fmaf(__int_as_float(e.y), in[e.x * DIM + lane], acc);
        }
    }
    out[row * DIM + lane] = acc;
}

// ---------------------------------------------------------------------------
// 5) h += step * (a2 - sigmoid(alpha_logit)*h + static/norm)   (float4)
// ---------------------------------------------------------------------------
__global__ __launch_bounds__(256) void update_kernel(float4* __restrict__ h4,
                                                     const float4* __restrict__ a4,
                                                     const float4* __restrict__ su4,
                                                     const float4* __restrict__ si4,
                                                     const float* __restrict__ alpha_logit,
                                                     const unsigned* __restrict__ maxsq_bits,
                                                     const float* __restrict__ dt) {
    int i = blockIdx.x * 256 + threadIdx.x;          // over N_NODES * 8 float4s
    if (i >= N_NODES * (DIM / 4)) return;
    const float inv  = 1.0f / sqrtf(__uint_as_float(*maxsq_bits));
    const float step = dt[0] * (1.0f / K_STEPS);
    const int row = i >> 3;                          // 8 float4s per row
    const float alpha = 1.0f / (1.0f + __expf(-alpha_logit[row]));
    float4 hv = h4[i];
    float4 av = a4[i];
    float4 sv = (row < N_USERS) ? su4[i] : si4[i - N_USERS * (DIM / 4)];
    float4 r;
    r.x = hv.x + step * (av.x - alpha * hv.x + sv.x * inv);
    r.y = hv.y + step * (av.y - alpha * hv.y + sv.y * inv);
    r.z = hv.z + step * (av.z - alpha * hv.z + sv.z * inv);
    r.w = hv.w + step * (av.w - alpha * hv.w + sv.w * inv);
    h4[i] = r;
}

// ---------------------------------------------------------------------------
extern "C" void kernel_launch(void* const* d_in, const int* in_sizes, int n_in,
                              void* d_out, int out_size, void* d_ws, size_t ws_size,
                              hipStream_t stream) {
    const float* xu          = (const float*)d_in[0];
    const float* xi          = (const float*)d_in[1];
    const float* static_u    = (const float*)d_in[2];
    const float* static_i    = (const float*)d_in[3];
    const float* edge_w      = (const float*)d_in[4];
    const float* alpha_logit = (const float*)d_in[5];
    const float* dt          = (const float*)d_in[6];
    const int*   edge_src    = (const int*)d_in[7];
    const int*   edge_dst    = (const int*)d_in[8];

    // workspace carve-up (256 B aligned regions)
    char* p = (char*)d_ws;
    auto alloc = [&](size_t bytes) {
        void* r = (void*)p;
        p += (bytes + 255) & ~(size_t)255;
        return r;
    };
    float*    h       = (float*)alloc((size_t)N_NODES * DIM * sizeof(float));
    float*    t1      = (float*)alloc((size_t)N_NODES * DIM * sizeof(float));
    float*    t2      = (float*)alloc((size_t)N_NODES * DIM * sizeof(float));
    int2*     ep      = (int2*)alloc((size_t)N_EDGES * sizeof(int2));
    int*      row_ptr = (int*)alloc((size_t)(N_NODES + 1) * sizeof(int));
    int*      cursor  = (int*)alloc((size_t)N_NODES * sizeof(int));
    int*      counts  = (int*)alloc((size_t)N_NODES * sizeof(int));
    int*      partial = (int*)alloc(1024 * sizeof(int));
    unsigned* maxsq   = (unsigned*)alloc(256);

    const int nodeBlocks  = (N_NODES + 7) / 8;                 // 1 wave per row
    const int edgeBlocks  = (N_EDGES + 255) / 256;
    const int scanBlocks  = (N_NODES + 255) / 256;             // 586
    const int vecBlocks   = (N_NODES * (DIM / 4) + 255) / 256; // float4 grids

    (void)hipMemsetAsync(maxsq, 0, sizeof(unsigned), stream);
    (void)hipMemsetAsync(counts, 0, (size_t)N_NODES * sizeof(int), stream);

    norm_kernel<<<nodeBlocks, 256, 0, stream>>>(xu, xi, maxsq);
    count_kernel<<<edgeBlocks, 256, 0, stream>>>(edge_dst, counts);
    scan1_kernel<<<scanBlocks, 256, 0, stream>>>(counts, row_ptr, partial, N_NODES);
    scan2_kernel<<<1, 1024, 0, stream>>>(partial, scanBlocks);
    scan3_kernel<<<scanBlocks, 256, 0, stream>>>(row_ptr, cursor, partial, N_NODES);
    fill_kernel<<<edgeBlocks, 256, 0, stream>>>(edge_src, edge_dst, edge_w, cursor, ep);

    init_h_kernel<<<vecBlocks, 256, 0, stream>>>((const float4*)xu, (const float4*)xi,
                                                 (float4*)h, maxsq);

    for (int k = 0; k < K_STEPS; ++k) {
        prop_kernel<<<nodeBlocks, 256, 0, stream>>>(h, t1, ep, row_ptr);
        prop_kernel<<<nodeBlocks, 256, 0, stream>>>(t1, t2, ep, row_ptr);
        update_kernel<<<vecBlocks, 256, 0, stream>>>((float4*)h, (const float4*)t2,
                                                     (const float4*)static_u,
                                                     (const float4*)static_i,
                                                     alpha_logit, maxsq, dt);
    }

    (void)hipMemcpyAsync(d_out, h, (size_t)N_NODES * DIM * sizeof(float),
                         hipMemcpyDeviceToDevice, stream);
}